// MultiRectSDF_47141561041406
// MI455X (gfx1250) — compile-verified
//
#include <hip/hip_runtime.h>
#include <math.h>

typedef float v2f __attribute__((ext_vector_type(2)));
typedef float v8f __attribute__((ext_vector_type(8)));

// Rect SDF for one (point, shape) pair given rotated+translated coords.
// sq >= 0 always and sqrt(0) == 0, so the reference's (sq>0) guard reduces
// to a plain sqrt; use the raw v_sqrt_f32 (<=1 ulp).
__device__ __forceinline__ float rect_sdf(float px, float py, float rx, float ry) {
    float dx = __builtin_fabsf(px) - rx;
    float dy = __builtin_fabsf(py) - ry;
    float mx = fmaxf(dx, 0.0f);
    float my = fmaxf(dy, 0.0f);
    float sq = __builtin_fmaf(my, my, mx * mx);
    float outside = __builtin_amdgcn_sqrtf(sq);
    float inside  = fminf(fmaxf(dx, dy), 0.0f);
    return outside + inside;
}

__global__ __launch_bounds__(256) void
MultiRectSDF_wmma_kernel(const float2* __restrict__ query,
                         const float2* __restrict__ trans,
                         const float2* __restrict__ rads,
                         const float*  __restrict__ angles,
                         float* __restrict__ out,
                         int npoints, int ntiles) {
    const int lane          = threadIdx.x & 31;
    const int waveInBlock   = threadIdx.x >> 5;
    const int wavesPerBlock = blockDim.x >> 5;
    const int globalWave    = blockIdx.x * wavesPerBlock + waveInBlock;
    const int totalWaves    = gridDim.x * wavesPerBlock;

    const int  m  = lane & 15;   // A-row shape index within half / B-col point index
    const bool lo = lane < 16;

    // ---- Per-wave constant setup: A matrices (shape coefficients) ----
    // A is 16x4 (M=shape, K=coeff). Lanes 0-15 hold K=0,1 of row M=lane;
    // lanes 16-31 hold K=2,3 of row M=lane-16.
    // x-row of shape s: (c, -s, -tx, 0); y-row: (s, c, -ty, 0).
    float  ang0 = angles[m];
    float  ang1 = angles[m + 16];
    float2 t0   = trans[m];
    float2 t1   = trans[m + 16];
    float  c0 = __builtin_cosf(ang0), s0 = __builtin_sinf(ang0);
    float  c1 = __builtin_cosf(ang1), s1 = __builtin_sinf(ang1);

    v2f a_rx0, a_ry0, a_rx1, a_ry1;
    a_rx0.x = lo ?  c0 : -t0.x;  a_rx0.y = lo ? -s0 : 0.0f;
    a_ry0.x = lo ?  s0 : -t0.y;  a_ry0.y = lo ?  c0 : 0.0f;
    a_rx1.x = lo ?  c1 : -t1.x;  a_rx1.y = lo ? -s1 : 0.0f;
    a_ry1.x = lo ?  s1 : -t1.y;  a_ry1.y = lo ?  c1 : 0.0f;

    // Per-lane rads: D row r (VGPR r) corresponds to shape rbase+r (group 0)
    // and 16+rbase+r (group 1), where rbase = 0 for lanes 0-15, 8 for 16-31.
    const int rbase = lo ? 0 : 8;
    float2 rad0[8], rad1[8];
#pragma unroll
    for (int r = 0; r < 8; ++r) {
        rad0[r] = rads[rbase + r];
        rad1[r] = rads[16 + rbase + r];
    }

    const float bsel_x = lo ? 0.0f : 1.0f;   // homogeneous constants for hi half
    const int   pmax   = npoints - 1;

    // ---- Grid-stride loop over 16-point tiles ----
    for (int tile = globalWave; tile < ntiles; tile += totalWaves) {
        const int base = tile * 16;
        const int p    = base + m;

        // B is 4x16 (K=coeff, N=point): column p = (qx, qy, 1, 0).
        // Lanes 0-15: VGPR0=K0(qx), VGPR1=K1(qy). Lanes 16-31: VGPR0=K2(1), VGPR1=K3(0).
        // Branchless: all lanes load (hi half hits the same cachelines), then
        // two cndmasks splice in the (1,0) homogeneous row for lanes 16-31.
        const int pc = (p < pmax) ? p : pmax;
        float2 q = query[pc];
        v2f b;
        b.x = lo ? q.x : bsel_x;   // hi: 1.0
        b.y = lo ? q.y : 0.0f;     // hi: 0.0

        v8f acc = {};
        // D = A x B : 16 shapes x 16 points, f32 accumulate from zero.
        v8f d_rx0 = __builtin_amdgcn_wmma_f32_16x16x4_f32(false, a_rx0, false, b, (short)0, acc, false, false);
        v8f d_ry0 = __builtin_amdgcn_wmma_f32_16x16x4_f32(false, a_ry0, false, b, (short)0, acc, false, false);
        v8f d_rx1 = __builtin_amdgcn_wmma_f32_16x16x4_f32(false, a_rx1, false, b, (short)0, acc, false, false);
        v8f d_ry1 = __builtin_amdgcn_wmma_f32_16x16x4_f32(false, a_ry1, false, b, (short)0, acc, false, false);

        // Elementwise rect SDF + in-lane min over this lane's 16 shapes.
        float best = rect_sdf(d_rx0[0], d_ry0[0], rad0[0].x, rad0[0].y);
        best = fminf(best, rect_sdf(d_rx1[0], d_ry1[0], rad1[0].x, rad1[0].y));
#pragma unroll
        for (int r = 1; r < 8; ++r) {
            best = fminf(best, rect_sdf(d_rx0[r], d_ry0[r], rad0[r].x, rad0[r].y));
            best = fminf(best, rect_sdf(d_rx1[r], d_ry1[r], rad1[r].x, rad1[r].y));
        }

        // Merge the two shape-halves: lane l <-> lane l^16 (ds_swizzle SWAPX16:
        // group-of-32 mode, xor=0x10, and=0x1f -> offset 0x401f).
        int sw = __builtin_amdgcn_ds_swizzle(__float_as_int(best), 0x401f);
        best = fminf(best, __int_as_float(sw));

        if (lo && p < npoints) out[p] = best;   // coalesced 16-lane store
    }
}

extern "C" void kernel_launch(void* const* d_in, const int* in_sizes, int n_in,
                              void* d_out, int out_size, void* d_ws, size_t ws_size,
                              hipStream_t stream) {
    (void)n_in; (void)d_ws; (void)ws_size;
    const float2* query  = (const float2*)d_in[0];
    const float2* trans  = (const float2*)d_in[1];
    const float2* rads   = (const float2*)d_in[2];
    const float*  angles = (const float*)d_in[3];
    float* out = (float*)d_out;

    const int npoints = in_sizes[0] / 2;     // query is (N,2)
    const int ntiles  = (npoints + 15) / 16;

    const int block = 256;                   // 8 waves per block (wave32)
    int grid = 2048;                         // ~16384 waves, ~8 tiles/wave
    const int wavesPerBlock = block / 32;
    if (grid * wavesPerBlock > ntiles)
        grid = (ntiles + wavesPerBlock - 1) / wavesPerBlock;
    if (grid < 1) grid = 1;

    MultiRectSDF_wmma_kernel<<<grid, block, 0, stream>>>(
        query, trans, rads, angles, out, npoints, ntiles);
}